// if_encoder_14946486190132
// MI455X (gfx1250) — compile-verified
//
#include <hip/hip_runtime.h>
#include <stdint.h>

// Problem constants (from the reference)
#define B_      64
#define D_      4096
#define T_      128
#define NCH     (B_ * D_)        // 262144 chains
#define BLOCK   256              // 8 wave32s
#define TCH     16               // time-steps per chunk
#define NCHUNKS (T_ / TCH)       // 8
#define ROWF    (TCH + 4)        // 20 floats = 80 bytes: 16B aligned, bank-skewed

// The async-LDS builtins take pointers to a 16-byte int vector.
typedef int v4i __attribute__((vector_size(16)));
typedef __attribute__((address_space(1))) v4i gv4i;  // global int4
typedef __attribute__((address_space(3))) v4i lv4i;  // LDS int4

// ---- CDNA5 async global<->LDS helpers (ASYNCcnt-tracked) ----------------

__device__ __forceinline__ void async_ld_b128(void* lds, const void* g) {
#if __has_builtin(__builtin_amdgcn_global_load_async_to_lds_b128)
  __builtin_amdgcn_global_load_async_to_lds_b128((gv4i*)g, (lv4i*)lds, 0, 0);
#else
  asm volatile("global_load_async_to_lds_b128 %0, %1, off"
               :: "v"(lds), "v"(g) : "memory");
#endif
}

__device__ __forceinline__ void async_st_b128(void* g, void* lds) {
#if __has_builtin(__builtin_amdgcn_global_store_async_from_lds_b128)
  __builtin_amdgcn_global_store_async_from_lds_b128((gv4i*)g, (lv4i*)lds, 0, 0);
#else
  asm volatile("global_store_async_from_lds_b128 %0, %1, off"
               :: "v"(g), "v"(lds) : "memory");
#endif
}

__device__ __forceinline__ void wait_async0() {
#if __has_builtin(__builtin_amdgcn_s_wait_asynccnt)
  __builtin_amdgcn_s_wait_asynccnt(0);
#else
  asm volatile("s_wait_asynccnt 0" ::: "memory");
#endif
}

// ---- Kernel -------------------------------------------------------------
// One thread = one (b,d) chain; membrane potential v stays in a register.
// Double-buffered async LDS staging of 16-step time tiles; spikes written
// back in place and streamed out with async stores from LDS.

__global__ __launch_bounds__(BLOCK)
void if_encoder_scan(const float* __restrict__ in,     // [B,D,T]
                     const float* __restrict__ v0,     // [B,D]
                     const float* __restrict__ thr_p,  // scalar
                     float* __restrict__ spikes,       // [B,D,T]
                     float* __restrict__ vout)         // [B,D]
{
  __shared__ __align__(16) float buf[2][BLOCK * ROWF];   // 2 x 20 KB

  const int    tid    = threadIdx.x;
  const size_t chain0 = (size_t)blockIdx.x * BLOCK;
  const int    word   = tid & 3;   // which 16B word of a 64B row-segment
  const int    cpart  = tid >> 2;  // 0..63: chain sub-index within a round

  const float thr = *thr_p;

  // Prologue: async-load chunk 0 (each wave instruction covers 8 full
  // 64B row-segments -> cacheline-granular global requests).
#pragma unroll
  for (int r = 0; r < 4; ++r) {
    const int cl = r * 64 + cpart;
    async_ld_b128(&buf[0][cl * ROWF + word * 4],
                  in + (chain0 + cl) * T_ + word * 4);
  }

  float v = v0[chain0 + tid];

  for (int k = 0; k < NCHUNKS; ++k) {
    // All of this wave's outstanding async ops done (loads of chunk k,
    // stores of chunk k-1); barrier extends that to the whole block so the
    // buffer we are about to refill is no longer being read by stores.
    wait_async0();
    __syncthreads();

    // Prefetch chunk k+1 into the other buffer (overlaps with compute).
    if (k + 1 < NCHUNKS) {
#pragma unroll
      for (int r = 0; r < 4; ++r) {
        const int cl = r * 64 + cpart;
        async_ld_b128(&buf[(k + 1) & 1][cl * ROWF + word * 4],
                      in + (chain0 + cl) * T_ + (size_t)(k + 1) * TCH + word * 4);
      }
    }

    // Scan 16 steps of this thread's own chain; overwrite inputs with
    // spikes in place (thread-local row -> no cross-thread hazard).
    float* row = &buf[k & 1][tid * ROWF];
#pragma unroll
    for (int q = 0; q < TCH / 4; ++q) {
      float4 x = *reinterpret_cast<float4*>(row + q * 4);  // ds_load_b128
      float4 s;
      v += x.x; s.x = (v >= thr) ? 1.0f : 0.0f; v -= s.x * thr;
      v += x.y; s.y = (v >= thr) ? 1.0f : 0.0f; v -= s.y * thr;
      v += x.z; s.z = (v >= thr) ? 1.0f : 0.0f; v -= s.z * thr;
      v += x.w; s.w = (v >= thr) ? 1.0f : 0.0f; v -= s.w * thr;
      *reinterpret_cast<float4*>(row + q * 4) = s;         // ds_store_b128
    }
    __syncthreads();  // spike tile fully in LDS before async stores read it

    // Stream the spike tile out (LDS -> global, ASYNCcnt-tracked).
#pragma unroll
    for (int r = 0; r < 4; ++r) {
      const int cl = r * 64 + cpart;
      async_st_b128(spikes + (chain0 + cl) * T_ + (size_t)k * TCH + word * 4,
                    &buf[k & 1][cl * ROWF + word * 4]);
    }
  }

  // Final membrane potential, coalesced. (S_ENDPGM's implicit wait-idle
  // covers the last async stores.)
  vout[chain0 + tid] = v;
}

// ---- Harness entry ------------------------------------------------------

extern "C" void kernel_launch(void* const* d_in, const int* in_sizes, int n_in,
                              void* d_out, int out_size, void* d_ws, size_t ws_size,
                              hipStream_t stream) {
  (void)in_sizes; (void)n_in; (void)out_size; (void)d_ws; (void)ws_size;

  const float* in    = (const float*)d_in[0];  // input_spikes [64,4096,128] f32
  const float* v0    = (const float*)d_in[1];  // states       [64,4096]     f32
  const float* thr   = (const float*)d_in[2];  // threshold scalar           f32

  float* out_spikes = (float*)d_out;                        // [64,4096,128]
  float* out_v      = (float*)d_out + (size_t)NCH * T_;     // [64,4096]

  dim3 grid(NCH / BLOCK);  // 1024 blocks
  dim3 block(BLOCK);
  if_encoder_scan<<<grid, block, 0, stream>>>(in, v0, thr, out_spikes, out_v);
}